// HyperConv_31061203485206
// MI455X (gfx1250) — compile-verified
//
#include <hip/hip_runtime.h>

typedef __attribute__((ext_vector_type(2))) float v2f;
typedef __attribute__((ext_vector_type(8))) float v8f;

#define BATCH   8
#define GNUM    4
#define CHIN    256
#define CHOUT   256
#define HYPIN   128
#define GIN     64
#define GOUT    64
#define NW      36864           // GOUT*GIN*9
#define NB      64
#define NWNB    36928           // NW+NB
#define NWNB4   9232            // NWNB/4
#define HD      2324
#define HH      64
#define WWIDTH  64
#define KTOT    576             // GIN*9
#define ROWS    4               // output rows per block
#define TROWS   6               // ROWS + 2 halo
#define XSTR    68              // padded LDS column stride (66 used)

#if __has_builtin(__builtin_amdgcn_global_load_async_to_lds_b32)
#define HAVE_ASYNC_LDS 1
typedef __attribute__((address_space(1))) int* gint_p;
typedef __attribute__((address_space(3))) int* lint_p;
__device__ __forceinline__ void async_copy_b32(const float* gsrc, float* ldst) {
    __builtin_amdgcn_global_load_async_to_lds_b32(
        (gint_p)gsrc, (lint_p)ldst, 0, 0);
}
#endif

// ---------------- Phase A1: h[g,b,d] = relu(hyper[b,:] @ W1[g,:,d] + b1[g,d])
__global__ __launch_bounds__(256) void hyper_mlp1(
    const float* __restrict__ hyper, const float* __restrict__ W1,
    const float* __restrict__ b1, float* __restrict__ h)
{
    const int gb = blockIdx.x;
    const int g  = gb >> 3;
    const int b  = gb & 7;
    const int d  = blockIdx.y * 256 + threadIdx.x;

    __shared__ float hv[HYPIN];
    if (threadIdx.x < HYPIN) hv[threadIdx.x] = hyper[b * HYPIN + threadIdx.x];
    __syncthreads();

    if (d < HD) {
        const float* wp = W1 + (size_t)g * HYPIN * HD + d;
        float acc = b1[(size_t)g * HD + d];
        #pragma unroll 4
        for (int i = 0; i < HYPIN; ++i)
            acc += hv[i] * wp[(size_t)i * HD];
        h[(size_t)(g * BATCH + b) * HD + d] = fmaxf(acc, 0.f);
    }
}

// ---------------- Phase A2: gened[g,b,o] = h[g,b,:] @ W2[g,:,o] + b2[g,o]
// HBM stream over W2 (1.37 GB, read exactly once) with float4 loads:
// 16 B -> 32 FMAs per VMEM instruction.  h broadcast from LDS chunks.
__global__ __launch_bounds__(256) void hyper_mlp2(
    const float* __restrict__ h, const float* __restrict__ W2,
    const float* __restrict__ b2, float* __restrict__ gened)
{
    __shared__ float hs[BATCH][260];
    const int g   = blockIdx.x;
    const int tid = threadIdx.x;
    const int o4  = blockIdx.y * 256 + tid;     // float4 column index
    const bool valid = o4 < NWNB4;

    float4 acc[BATCH];
    #pragma unroll
    for (int bb = 0; bb < BATCH; ++bb)
        acc[bb] = make_float4(0.f, 0.f, 0.f, 0.f);

    const float* hg = h + (size_t)g * BATCH * HD;
    const float4* w2g = (const float4*)(W2 + (size_t)g * HD * NWNB) + o4;

    for (int d0 = 0; d0 < HD; d0 += 256) {
        const int dlen = min(256, HD - d0);
        for (int e = tid; e < BATCH * 256; e += 256) {
            int bb = e >> 8;
            int dd = e & 255;
            if (dd < dlen) hs[bb][dd] = hg[(size_t)bb * HD + d0 + dd];
        }
        __syncthreads();
        if (valid) {
            const float4* wp = w2g + (size_t)d0 * NWNB4;
            for (int dd = 0; dd < dlen; ++dd) {
                float4 w = wp[(size_t)dd * NWNB4];
                #pragma unroll
                for (int bb = 0; bb < BATCH; ++bb) {
                    float hval = hs[bb][dd];
                    acc[bb].x += hval * w.x;
                    acc[bb].y += hval * w.y;
                    acc[bb].z += hval * w.z;
                    acc[bb].w += hval * w.w;
                }
            }
        }
        __syncthreads();
    }
    if (valid) {
        float4 bias = ((const float4*)(b2 + (size_t)g * NWNB))[o4];
        #pragma unroll
        for (int bb = 0; bb < BATCH; ++bb) {
            float4 r;
            r.x = acc[bb].x + bias.x;
            r.y = acc[bb].y + bias.y;
            r.z = acc[bb].z + bias.z;
            r.w = acc[bb].w + bias.w;
            ((float4*)(gened + (size_t)(g * BATCH + bb) * NWNB))[o4] = r;
        }
    }
}

// ---------------- Phase B: grouped dynamic conv as fp32 WMMA GEMM.
// Block = one (b,g) x 4 output rows.  LDS stages image tile laid out
// [row][c][x] so K-loop LDS deltas are constants.  Each wave runs 2 passes;
// a pass computes one (otile,row) with all 4 l-subtiles (4 accumulators)
// sharing a single A fragment: per K-step = 1 b64 + 8 ds + 4 WMMA.
__global__ __launch_bounds__(256) void dynconv_wmma(
    const float* __restrict__ image, const float* __restrict__ gened,
    float* __restrict__ out)
{
    extern __shared__ float img[];   // [TROWS][GIN][XSTR]
    const int bg  = blockIdx.x;
    const int b   = bg >> 2;
    const int g   = bg & 3;
    const int h0  = blockIdx.y * ROWS;
    const int tid = threadIdx.x;

    // stage image tile (rows h0-1..h0+4, cols -1..64, zero-padded)
    const float* ibase = image + (size_t)(b * CHIN + g * GIN) * HH * WWIDTH;
    for (int e = tid; e < GIN * TROWS * 66; e += 256) {
        int c    = e / (TROWS * 66);
        int rem  = e - c * (TROWS * 66);
        int r    = rem / 66;
        int x    = rem - r * 66;
        int gh   = h0 + r - 1;
        int gw   = x - 1;
        int lidx = (r * GIN + c) * XSTR + x;
        if (gh >= 0 && gh < HH && gw >= 0 && gw < WWIDTH) {
#if HAVE_ASYNC_LDS
            async_copy_b32(&ibase[((size_t)c * HH + gh) * WWIDTH + gw], &img[lidx]);
#else
            img[lidx] = ibase[((size_t)c * HH + gh) * WWIDTH + gw];
#endif
        } else {
            img[lidx] = 0.f;
        }
    }
#if HAVE_ASYNC_LDS
    asm volatile("s_wait_asynccnt 0x0" ::: "memory");
#endif
    __syncthreads();

    const int wave = tid >> 5;
    const int lane = tid & 31;
    const int n    = lane & 15;          // M for A rows, N for B cols
    const int kb   = (lane >> 4) << 1;   // K sub-offset: 0 or 2
    const int hi   = (lane >> 4) << 3;   // C/D row offset: 0 or 8

    const float* Wg = gened + (size_t)(g * BATCH + b) * NWNB;

    for (int p = 0; p < 2; ++p) {
        const int sub   = wave * 8 + p * 4;   // 4 consecutive subtiles share otile,row
        const int otile = sub >> 4;           // 0..3
        const int row   = (sub & 15) >> 2;    // 0..3

        // Per-lane LDS float offsets for the 9-step macro period (36 k).
        // k = 36q + 4m + kb + {0,1}; c = k/9, tap = k%9, i = tap/3, j = tap%3
        // off(k) = ((row+i)*GIN + c)*XSTR + n + j ; q contributes +4*XSTR.
        int off0[9], off1[9];
        #pragma unroll
        for (int m = 0; m < 9; ++m) {
            int r0 = 4 * m + kb;
            int c0 = r0 / 9, t0 = r0 % 9;
            int i0 = t0 / 3, j0 = t0 % 3;
            off0[m] = ((row + i0) * GIN + c0) * XSTR + n + j0;
            int r1 = r0 + 1;
            int c1 = r1 / 9, t1 = r1 % 9;
            int i1 = t1 / 3, j1 = t1 % 3;
            off1[m] = ((row + i1) * GIN + c1) * XSTR + n + j1;
        }

        const float* wp = Wg + (size_t)(otile * 16 + n) * KTOT + kb;

        v8f acc[4];
        acc[0] = (v8f){}; acc[1] = (v8f){}; acc[2] = (v8f){}; acc[3] = (v8f){};

        for (int q = 0; q < 16; ++q) {
            #pragma unroll
            for (int m = 0; m < 9; ++m) {
                v2f a = *(const v2f*)(wp + m * 4);   // A: K={base,base+1}/{+2,+3}
                #pragma unroll
                for (int wq = 0; wq < 4; ++wq) {
                    v2f bf;
                    bf.x = img[off0[m] + wq * 16];
                    bf.y = img[off1[m] + wq * 16];
                    acc[wq] = __builtin_amdgcn_wmma_f32_16x16x4_f32(
                        false, a, false, bf, (short)0, acc[wq], false, false);
                }
            }
            wp += 36;
            #pragma unroll
            for (int m = 0; m < 9; ++m) { off0[m] += 4 * XSTR; off1[m] += 4 * XSTR; }
        }

        // store: VGPR v -> (M = v + hi, N = n); add generated bias
        float* outp = out + ((size_t)(b * CHOUT + g * GOUT) * HH + (h0 + row)) * WWIDTH + n;
        #pragma unroll
        for (int v = 0; v < 8; ++v) {
            int o = otile * 16 + hi + v;
            float bias = Wg[NW + o];
            float* po = outp + (size_t)o * (HH * WWIDTH);
            po[0]  = acc[0][v] + bias;
            po[16] = acc[1][v] + bias;
            po[32] = acc[2][v] + bias;
            po[48] = acc[3][v] + bias;
        }
    }
}

extern "C" void kernel_launch(void* const* d_in, const int* in_sizes, int n_in,
                              void* d_out, int out_size, void* d_ws, size_t ws_size,
                              hipStream_t stream) {
    const float* image = (const float*)d_in[0];
    const float* hyper = (const float*)d_in[1];
    const float* W1    = (const float*)d_in[2];
    const float* b1    = (const float*)d_in[3];
    const float* W2    = (const float*)d_in[4];
    const float* b2    = (const float*)d_in[5];
    float* out   = (float*)d_out;
    float* hbuf  = (float*)d_ws;                           // G*B*HD floats
    float* gened = hbuf + (size_t)GNUM * BATCH * HD;       // G*B*NWNB floats

    hyper_mlp1<<<dim3(GNUM * BATCH, (HD + 255) / 256), 256, 0, stream>>>(
        hyper, W1, b1, hbuf);
    hyper_mlp2<<<dim3(GNUM, (NWNB4 + 255) / 256), 256, 0, stream>>>(
        hbuf, W2, b2, gened);

    size_t shmem = (size_t)TROWS * GIN * XSTR * sizeof(float); // 104448 B
    dynconv_wmma<<<dim3(BATCH * GNUM, HH / ROWS), 256, shmem, stream>>>(
        image, gened, out);
}